// NeighborlistForInference_56521769615745
// MI455X (gfx1250) — compile-verified
//
#include <hip/hip_runtime.h>
#include <stdint.h>

// NeighborlistForInference — brute N^2 pair list, MI455X (gfx1250).
// Store-bandwidth-bound: 32 B out/pair * 18M pairs = 576 MB vs 0.36 GFLOP
// (AI ~0.6 flop/B -> floor ~25 us at 23.3 TB/s). CDNA5 paths used:
//   * GLOBAL_LOAD_ASYNC_TO_LDS_B128 staging of the 72 KB position array
//     (ASYNCcnt + s_wait_asynccnt), all pair reads hit LDS conflict-free.
//   * Non-temporal aligned b128 stores (output is 3x the 192 MB L2).
// WMMA inapplicable: PBC wrap breaks the bilinear distance decomposition and
// compute is ~1000x under the BW roofline.

#define NMAX    6000
#define CUTOFF  0.5f

#define AS1 __attribute__((address_space(1)))
#define AS3 __attribute__((address_space(3)))

typedef __attribute__((ext_vector_type(2))) long long v2i64;
typedef __attribute__((ext_vector_type(4))) float     v4f;
typedef int v4i32 __attribute__((vector_size(16)));   // matches builtin proto V4i

__device__ __forceinline__ void async_copy16(const float* g, float* l) {
#if __has_builtin(__builtin_amdgcn_global_load_async_to_lds_b128)
  // proto: void (v4i32 AS1*, v4i32 AS3*, imm int offset, imm int cpol)
  __builtin_amdgcn_global_load_async_to_lds_b128(
      (AS1 v4i32*)g, (AS3 v4i32*)l, /*offset=*/0, /*cpol=*/0);
#else
  unsigned ldsoff = (unsigned)(unsigned long long)(AS3 float*)l;
  asm volatile("global_load_async_to_lds_b128 %0, %1, off"
               :: "v"(ldsoff), "v"(g) : "memory");
#endif
}

__device__ __forceinline__ void wait_async0() {
#if __has_builtin(__builtin_amdgcn_s_wait_asynccnt)
  __builtin_amdgcn_s_wait_asynccnt(0);
#else
  asm volatile("s_wait_asynccnt 0" ::: "memory");
#endif
}

// Faithful to remainder(x + L/2, L) - L/2 for x in (-L, L): t = x+h lands in
// (-L/2, 3L/2), so floor(t/L) is in {-1,0,1}. fmod is exact for |t| < L and
// t-L is exact for t in [L, 2L), so the staged select form reproduces the
// jnp.remainder arithmetic (including the +h/-h rounding round-trip) without
// a per-pair f32 division.
__device__ __forceinline__ float wrap1(float x, float L, float h) {
  float t = x + h;
  t = (t >= L)   ? (t - L) : t;
  t = (t < 0.0f) ? (t + L) : t;
  return t - h;
}

extern "C" __global__ void __launch_bounds__(256)
nbl_kernel(const float* __restrict__ pos, const float* __restrict__ box,
           const int* __restrict__ isper,
           long long* __restrict__ idx0, long long* __restrict__ idx1,
           float* __restrict__ dout, float* __restrict__ rout, int n)
{
  __shared__ float s_pos[3 * NMAX];            // 72000 B: whole position array
  const int tid   = threadIdx.x;
  const int total = 3 * n;

  // ---- Stage positions into LDS with async b128 copies (ASYNCcnt) ----
  const int nchunk16 = total >> 2;             // 16-byte chunks
  for (int c = tid; c < nchunk16; c += 256)
    async_copy16(pos + 4 * c, s_pos + 4 * c);
  for (int t = (nchunk16 << 2) + tid; t < total; t += 256)
    s_pos[t] = pos[t];                         // tail (none for n=6000)
  wait_async0();
  __syncthreads();

  const float Lx = box[0], Ly = box[4], Lz = box[8];
  const float hx = 0.5f * Lx, hy = 0.5f * Ly, hz = 0.5f * Lz;
  const int periodic = isper[0];

  // Row pairing for perfect balance: block b handles rows b and (n-2-b),
  // total pairs per block == n.
  const int lastrow = n - 2;
  for (int rr = 0; rr < 2; ++rr) {
    const int i = (rr == 0) ? (int)blockIdx.x : lastrow - (int)blockIdx.x;
    if (rr == 1 && i <= (int)blockIdx.x) break;   // middle row (odd count) once
    const int len = n - 1 - i;
    const long long base = (long long)i * (n - 1) - ((long long)i * (i - 1)) / 2;

    const float xi = s_pos[3 * i], yi = s_pos[3 * i + 1], zi = s_pos[3 * i + 2];

    // Peel to p % 4 == 0 so idx(8B), d(4B), r(12B) packs are all 16B-aligned.
    int peel = (int)((4 - (base & 3)) & 3);
    if (peel > len) peel = len;
    const int packs = (len - peel) >> 2;

    // ---- aligned 4-wide fast path: all stores are NT b128 ----
    for (int c = tid; c < packs; c += 256) {
      const int e0 = peel + 4 * c;
      const int j0 = i + 1 + e0;
      float rx[4], ry[4], rz[4], dd[4];
      int   inq[4];
#pragma unroll
      for (int q = 0; q < 4; ++q) {
        const int j = j0 + q;
        float dx = xi - s_pos[3 * j];
        float dy = yi - s_pos[3 * j + 1];
        float dz = zi - s_pos[3 * j + 2];
        if (periodic) {
          dx = wrap1(dx, Lx, hx);
          dy = wrap1(dy, Ly, hy);
          dz = wrap1(dz, Lz, hz);
        }
        const float d = sqrtf(dx * dx + dy * dy + dz * dz);
        const int in = d <= CUTOFF;
        inq[q] = in;
        dd[q]  = in ? d  : 0.0f;
        rx[q]  = in ? dx : 0.0f;
        ry[q]  = in ? dy : 0.0f;
        rz[q]  = in ? dz : 0.0f;
      }
      const long long p = base + e0;            // p % 4 == 0

      v2i64 ia, ib, ja, jb;
      ia.x = inq[0] ? (long long)i : -1LL;  ia.y = inq[1] ? (long long)i : -1LL;
      ib.x = inq[2] ? (long long)i : -1LL;  ib.y = inq[3] ? (long long)i : -1LL;
      ja.x = inq[0] ? (long long)(j0)     : -1LL;
      ja.y = inq[1] ? (long long)(j0 + 1) : -1LL;
      jb.x = inq[2] ? (long long)(j0 + 2) : -1LL;
      jb.y = inq[3] ? (long long)(j0 + 3) : -1LL;
      __builtin_nontemporal_store(ia, (v2i64*)(idx0 + p));
      __builtin_nontemporal_store(ib, (v2i64*)(idx0 + p + 2));
      __builtin_nontemporal_store(ja, (v2i64*)(idx1 + p));
      __builtin_nontemporal_store(jb, (v2i64*)(idx1 + p + 2));

      v4f dv = { dd[0], dd[1], dd[2], dd[3] };
      __builtin_nontemporal_store(dv, (v4f*)(dout + p));

      v4f r0 = { rx[0], ry[0], rz[0], rx[1] };
      v4f r1 = { ry[1], rz[1], rx[2], ry[2] };
      v4f r2 = { rz[2], rx[3], ry[3], rz[3] };
      float* rp = rout + 3 * p;                 // 12p bytes, 16B-aligned
      __builtin_nontemporal_store(r0, (v4f*)(rp));
      __builtin_nontemporal_store(r1, (v4f*)(rp + 4));
      __builtin_nontemporal_store(r2, (v4f*)(rp + 8));
    }

    // ---- scalar peel + tail ----
    for (int pass = 0; pass < 2; ++pass) {
      const int eb = (pass == 0) ? 0 : (peel + (packs << 2));
      const int ee = (pass == 0) ? peel : len;
      for (int e = eb + tid; e < ee; e += 256) {
        const int j = i + 1 + e;
        float dx = xi - s_pos[3 * j];
        float dy = yi - s_pos[3 * j + 1];
        float dz = zi - s_pos[3 * j + 2];
        if (periodic) {
          dx = wrap1(dx, Lx, hx);
          dy = wrap1(dy, Ly, hy);
          dz = wrap1(dz, Lz, hz);
        }
        const float d = sqrtf(dx * dx + dy * dy + dz * dz);
        const int in = d <= CUTOFF;
        const long long p = base + e;
        __builtin_nontemporal_store(in ? (long long)i : -1LL, idx0 + p);
        __builtin_nontemporal_store(in ? (long long)j : -1LL, idx1 + p);
        __builtin_nontemporal_store(in ? d : 0.0f, dout + p);
        float* rp = rout + 3 * p;
        __builtin_nontemporal_store(in ? dx : 0.0f, rp);
        __builtin_nontemporal_store(in ? dy : 0.0f, rp + 1);
        __builtin_nontemporal_store(in ? dz : 0.0f, rp + 2);
      }
    }
  }
}

extern "C" void kernel_launch(void* const* d_in, const int* in_sizes, int n_in,
                              void* d_out, int out_size, void* d_ws, size_t ws_size,
                              hipStream_t stream) {
  const float* pos   = (const float*)d_in[0];   // [n,3] f32
  const float* box   = (const float*)d_in[1];   // [3,3] f32
  const int*   isper = (const int*)d_in[2];     // scalar int
  const int n = in_sizes[0] / 3;                // 6000

  // Output layout (return-order concat): pair_indices [2,P] i64, d [P,1] f32, r [P,3] f32
  const long long P = (long long)n * (n - 1) / 2;
  long long* idx0 = (long long*)d_out;
  long long* idx1 = idx0 + P;
  float*     dptr = (float*)(idx1 + P);
  float*     rptr = dptr + P;

  const int rows = n - 1;
  const int nb   = (rows + 1) / 2;              // 3000 blocks, n pairs each
  nbl_kernel<<<nb, 256, 0, stream>>>(pos, box, isper, idx0, idx1, dptr, rptr, n);
  (void)d_ws; (void)ws_size; (void)n_in; (void)out_size;
}